// VAE_60438779789384
// MI455X (gfx1250) — compile-verified
//
#include <hip/hip_runtime.h>
#include <math.h>

#define V_SZ 32000
#define H_SZ 1024
#define L_SZ 256
#define C_SZ 32
#define T_SZ 64
#define G4H  (4 * H_SZ)
#define ZC   (L_SZ + C_SZ)
#define NB_REC 64              // blocks cooperating on the recurrence

typedef float v2f __attribute__((ext_vector_type(2)));
typedef float v8f __attribute__((ext_vector_type(8)));

__device__ __forceinline__ float sigmoidf(float x) { return 1.0f / (1.0f + __expf(-x)); }

__device__ __forceinline__ v8f wmma_f32(v2f a, v2f b, v8f c) {
  return __builtin_amdgcn_wmma_f32_16x16x4_f32(false, a, false, b, (short)0, c, false, false);
}

// ---------------------------------------------------------------------------
// K1: gather embeddings, build encoder h0, zero the recurrence barrier counters
// grid = T blocks, block = 256
// ---------------------------------------------------------------------------
__global__ void vae_prepare(const int* __restrict__ word, const int* __restrict__ tense,
                            const float* __restrict__ tense_emb,
                            const float* __restrict__ enc_emb, const float* __restrict__ dec_emb,
                            float* __restrict__ enc_x, float* __restrict__ dec_x,
                            float* __restrict__ h0enc, unsigned* __restrict__ cnts) {
  const int t = blockIdx.x;
  const int w = word[t];
  const int di = (t == 0) ? 0 : word[t - 1];   // SOS token = 0
  const int ts = tense[0];
  if (t == 0 && threadIdx.x < 2) cnts[threadIdx.x] = 0u;   // barrier counters
  for (int k = threadIdx.x; k < H_SZ; k += blockDim.x) {
    enc_x[t * H_SZ + k] = enc_emb[(size_t)w * H_SZ + k];
    float dv = dec_emb[(size_t)di * H_SZ + k];
    dec_x[t * H_SZ + k] = dv > 0.0f ? dv : 0.0f;          // relu
    if (t == 0)
      h0enc[k] = (k >= H_SZ - C_SZ) ? tense_emb[ts * C_SZ + (k - (H_SZ - C_SZ))] : 0.0f;
  }
}

// ---------------------------------------------------------------------------
// K2/K6: Y[64,N] = X[64,K] @ W[N,K]^T + bias0[n] (+ bias1[n]),  f32 WMMA 16x16x4.
// One wave per 32-column strip: 4 M-tiles x 2 N-tiles = eight 16x16 accumulators.
// Per k-step: 6 fragment loads feed 8 WMMAs; next iteration's fragments are
// prefetched into registers before the WMMA group (software pipeline) so the
// load latency overlaps the matrix ops. W is streamed exactly once.
// A frag (16x4 f32, 2 VGPR): lane<16 -> M=lane&15, K=k..k+1 ; lane>=16 -> K=k+2..k+3
// B frag (4x16 f32, 2 VGPR): same K split, N = lane&15  (B[k][n] = W[n][k])
// D:      elem r of lane -> row (r + 8*(lane>>4)), col (lane&15)
// ---------------------------------------------------------------------------
__global__ void wmma_gemm_xwt(const float* __restrict__ X, const float* __restrict__ W,
                              const float* __restrict__ bias0, const float* __restrict__ bias1,
                              float* __restrict__ Y, int N, int K) {
  const int lane  = threadIdx.x & 31;
  const int wave  = threadIdx.x >> 5;
  const int wpb   = blockDim.x >> 5;
  const int n0    = (blockIdx.x * wpb + wave) * 32;   // 32-column strip
  const int lm = lane & 15;
  const int hi = lane >> 4;

  const float* br0 = W + (size_t)(n0 + lm) * K + 2 * hi;
  const float* br1 = W + (size_t)(n0 + 16 + lm) * K + 2 * hi;
  const float* ar0 = X + (size_t)lm * K + 2 * hi;     // M rows 0..15
  const float* ar1 = ar0 + (size_t)16 * K;            // rows 16..31
  const float* ar2 = ar1 + (size_t)16 * K;            // rows 32..47
  const float* ar3 = ar2 + (size_t)16 * K;            // rows 48..63

  v8f z = {0.f,0.f,0.f,0.f,0.f,0.f,0.f,0.f};
  v8f acc00 = z, acc10 = z, acc20 = z, acc30 = z;     // x B-tile 0
  v8f acc01 = z, acc11 = z, acc21 = z, acc31 = z;     // x B-tile 1

  // pipeline prologue
  v2f b0 = *(const v2f*)br0, b1 = *(const v2f*)br1;
  v2f a0 = *(const v2f*)ar0, a1 = *(const v2f*)ar1;
  v2f a2 = *(const v2f*)ar2, a3 = *(const v2f*)ar3;

  for (int k = 4; k < K; k += 4) {
    // prefetch next fragments (overlaps with the 8 WMMAs below)
    v2f b0n = *(const v2f*)(br0 + k), b1n = *(const v2f*)(br1 + k);
    v2f a0n = *(const v2f*)(ar0 + k), a1n = *(const v2f*)(ar1 + k);
    v2f a2n = *(const v2f*)(ar2 + k), a3n = *(const v2f*)(ar3 + k);

    acc00 = wmma_f32(a0, b0, acc00);
    acc10 = wmma_f32(a1, b0, acc10);
    acc20 = wmma_f32(a2, b0, acc20);
    acc30 = wmma_f32(a3, b0, acc30);
    acc01 = wmma_f32(a0, b1, acc01);
    acc11 = wmma_f32(a1, b1, acc11);
    acc21 = wmma_f32(a2, b1, acc21);
    acc31 = wmma_f32(a3, b1, acc31);

    b0 = b0n; b1 = b1n; a0 = a0n; a1 = a1n; a2 = a2n; a3 = a3n;
  }
  // pipeline epilogue
  acc00 = wmma_f32(a0, b0, acc00);
  acc10 = wmma_f32(a1, b0, acc10);
  acc20 = wmma_f32(a2, b0, acc20);
  acc30 = wmma_f32(a3, b0, acc30);
  acc01 = wmma_f32(a0, b1, acc01);
  acc11 = wmma_f32(a1, b1, acc11);
  acc21 = wmma_f32(a2, b1, acc21);
  acc31 = wmma_f32(a3, b1, acc31);

  float bs0 = 0.0f, bs1 = 0.0f;
  if (bias0) { bs0 += bias0[n0 + lm]; bs1 += bias0[n0 + 16 + lm]; }
  if (bias1) { bs0 += bias1[n0 + lm]; bs1 += bias1[n0 + 16 + lm]; }
  const int c0 = n0 + lm;
  const int c1 = n0 + 16 + lm;
#pragma unroll
  for (int r = 0; r < 8; ++r) {
    const int row = r + 8 * hi;
    Y[(size_t)(row)      * N + c0] = acc00[r] + bs0;
    Y[(size_t)(row + 16) * N + c0] = acc10[r] + bs0;
    Y[(size_t)(row + 32) * N + c0] = acc20[r] + bs0;
    Y[(size_t)(row + 48) * N + c0] = acc30[r] + bs0;
    Y[(size_t)(row)      * N + c1] = acc01[r] + bs1;
    Y[(size_t)(row + 16) * N + c1] = acc11[r] + bs1;
    Y[(size_t)(row + 32) * N + c1] = acc21[r] + bs1;
    Y[(size_t)(row + 48) * N + c1] = acc31[r] + bs1;
  }
}

// ---------------------------------------------------------------------------
// K3/K5: distributed sequential LSTM. NB_REC blocks x 1024 threads.
// Block b owns hidden units [b*16, b*16+16). Thread layout:
//   unit = tid>>6 (16), gate = (tid>>4)&3 (4), s16 = tid&15 (16 k-chunks)
// Each thread keeps its 64 Whh weights in registers (Whh read from memory
// exactly once). Per step: broadcast h via LDS, 64 FMAs/thread, 16-lane
// shuffle reduce, 16 threads apply the cell, then a device-scope counter
// barrier (monotonic; zeroed by vae_prepare each launch).
// ---------------------------------------------------------------------------
__global__ void lstm_seq_dist(const float* __restrict__ Whh, const float* __restrict__ Xih,
                              const float* __restrict__ h0, float* __restrict__ h_g,
                              float* __restrict__ Hout, unsigned* __restrict__ cnt) {
  __shared__ float hs[H_SZ];
  __shared__ float gsum[16 * 4];
  const int tid  = threadIdx.x;
  const int b    = blockIdx.x;
  const int unit = tid >> 6;
  const int gate = (tid >> 4) & 3;
  const int s16  = tid & 15;
  const int kbase = s16 * 64;
  const int j = b * 16 + unit;

  // preload this thread's 64 weights: Whh[gate*H + j][kbase .. kbase+63]
  const float* wrow = Whh + (size_t)(gate * H_SZ + j) * H_SZ + kbase;
  float4 w[16];
#pragma unroll
  for (int i = 0; i < 16; ++i) w[i] = *(const float4*)(wrow + 4 * i);

  float c = 0.0f;   // cell state, meaningful for tid < 16
  for (int t = 0; t < T_SZ; ++t) {
    hs[tid] = (t == 0) ? h0[tid] : h_g[tid];
    __syncthreads();

    float p = 0.0f;
#pragma unroll
    for (int i = 0; i < 16; ++i) {
      float4 hv = *(const float4*)&hs[kbase + 4 * i];
      p += w[i].x * hv.x + w[i].y * hv.y + w[i].z * hv.z + w[i].w * hv.w;
    }
#pragma unroll
    for (int m = 8; m >= 1; m >>= 1) p += __shfl_xor(p, m, 16);
    if (s16 == 0) gsum[unit * 4 + gate] = p;
    __syncthreads();

    if (tid < 16) {
      const float* xg = Xih + (size_t)t * G4H;
      const int jj = b * 16 + tid;
      float gi = gsum[tid * 4 + 0] + xg[jj];
      float gf = gsum[tid * 4 + 1] + xg[jj + H_SZ];
      float gg = gsum[tid * 4 + 2] + xg[jj + 2 * H_SZ];
      float go = gsum[tid * 4 + 3] + xg[jj + 3 * H_SZ];
      c = sigmoidf(gf) * c + sigmoidf(gi) * tanhf(gg);
      float hn = sigmoidf(go) * tanhf(c);
      h_g[jj] = hn;
      if (Hout) Hout[(size_t)t * H_SZ + jj] = hn;
    }
    __threadfence();      // make this block's h writes visible device-wide
    __syncthreads();
    if (tid == 0) {
      __hip_atomic_fetch_add(cnt, 1u, __ATOMIC_RELEASE, __HIP_MEMORY_SCOPE_AGENT);
      const unsigned target = (unsigned)NB_REC * (unsigned)(t + 1);
      while (__hip_atomic_load(cnt, __ATOMIC_ACQUIRE, __HIP_MEMORY_SCOPE_AGENT) < target)
        __builtin_amdgcn_s_sleep(1);
    }
    __syncthreads();
  }
}

// ---------------------------------------------------------------------------
// K4a: mean/logvar/latent. 1 block, 256 threads (thread l -> latent dim l).
// ---------------------------------------------------------------------------
__global__ void latent_kernel(const float* __restrict__ h, const float* __restrict__ W_mean,
                              const float* __restrict__ b_mean, const float* __restrict__ W_var,
                              const float* __restrict__ b_var, const float* __restrict__ eps,
                              float* __restrict__ mean_out, float* __restrict__ logvar_out,
                              float* __restrict__ latent) {
  const int l = threadIdx.x;
  const float* wm = W_mean + (size_t)l * H_SZ;
  const float* wv = W_var + (size_t)l * H_SZ;
  float sm = b_mean[l], sv = b_var[l];
  for (int k = 0; k < H_SZ; k += 4) {
    float4 hv = *(const float4*)&h[k];
    float4 a = *(const float4*)&wm[k];
    float4 b = *(const float4*)&wv[k];
    sm += a.x * hv.x + a.y * hv.y + a.z * hv.z + a.w * hv.w;
    sv += b.x * hv.x + b.y * hv.y + b.z * hv.z + b.w * hv.w;
  }
  mean_out[l] = sm;
  logvar_out[l] = sv;
  latent[l] = sm + eps[l] * __expf(0.5f * sv);
}

// ---------------------------------------------------------------------------
// K4b: dec_h0 = [latent | tense_e] @ W_l2h^T + b_l2h. 1 block, 1024 threads.
// ---------------------------------------------------------------------------
__global__ void dec_init(const float* __restrict__ latent, const float* __restrict__ tense_emb,
                         const int* __restrict__ tense, const float* __restrict__ W_l2h,
                         const float* __restrict__ b_l2h, float* __restrict__ dec_h0) {
  __shared__ float z[ZC];
  const int j = threadIdx.x;
  if (j < L_SZ) z[j] = latent[j];
  else if (j < ZC) z[j] = tense_emb[tense[0] * C_SZ + (j - L_SZ)];
  __syncthreads();
  const float* wr = W_l2h + (size_t)j * ZC;
  float s = b_l2h[j];
#pragma unroll 4
  for (int k = 0; k < ZC; ++k) s += wr[k] * z[k];
  dec_h0[j] = s;
}

// ---------------------------------------------------------------------------
// K7: per-row log_softmax + first-index argmax. grid = T, block = 1024.
// ---------------------------------------------------------------------------
__global__ void logsoftmax_argmax(const float* __restrict__ logits, float* __restrict__ dist,
                                  float* __restrict__ pred) {
  const int t = blockIdx.x;
  const float* row = logits + (size_t)t * V_SZ;
  __shared__ float redf[1024];
  __shared__ int redi[1024];

  float m = -INFINITY; int mi = 0;
  for (int k = threadIdx.x; k < V_SZ; k += 1024) {
    float v = row[k];
    if (v > m) { m = v; mi = k; }
  }
  redf[threadIdx.x] = m; redi[threadIdx.x] = mi;
  __syncthreads();
  for (int s = 512; s > 0; s >>= 1) {
    if (threadIdx.x < s) {
      float vo = redf[threadIdx.x + s]; int io = redi[threadIdx.x + s];
      if (vo > redf[threadIdx.x] || (vo == redf[threadIdx.x] && io < redi[threadIdx.x])) {
        redf[threadIdx.x] = vo; redi[threadIdx.x] = io;
      }
    }
    __syncthreads();
  }
  const float rmax = redf[0];
  const int rarg = redi[0];
  __syncthreads();

  float s = 0.0f;
  for (int k = threadIdx.x; k < V_SZ; k += 1024) s += __expf(row[k] - rmax);
  redf[threadIdx.x] = s;
  __syncthreads();
  for (int st = 512; st > 0; st >>= 1) {
    if (threadIdx.x < st) redf[threadIdx.x] += redf[threadIdx.x + st];
    __syncthreads();
  }
  const float lse = __logf(redf[0]);
  for (int k = threadIdx.x; k < V_SZ; k += 1024)
    dist[(size_t)t * V_SZ + k] = row[k] - rmax - lse;
  if (threadIdx.x == 0) pred[t] = (float)rarg;
}

// ---------------------------------------------------------------------------
extern "C" void kernel_launch(void* const* d_in, const int* in_sizes, int n_in,
                              void* d_out, int out_size, void* d_ws, size_t ws_size,
                              hipStream_t stream) {
  const int*   word      = (const int*)d_in[0];
  const int*   tense     = (const int*)d_in[1];
  /* d_in[2] = use_teacher_forcing (unused by reference body) */
  const float* tense_emb = (const float*)d_in[3];
  const float* enc_emb   = (const float*)d_in[4];
  const float* enc_Wih   = (const float*)d_in[5];
  const float* enc_Whh   = (const float*)d_in[6];
  const float* enc_bih   = (const float*)d_in[7];
  const float* enc_bhh   = (const float*)d_in[8];
  const float* W_mean    = (const float*)d_in[9];
  const float* b_mean    = (const float*)d_in[10];
  const float* W_var     = (const float*)d_in[11];
  const float* b_var     = (const float*)d_in[12];
  const float* W_l2h     = (const float*)d_in[13];
  const float* b_l2h     = (const float*)d_in[14];
  const float* dec_emb   = (const float*)d_in[15];
  const float* dec_Wih   = (const float*)d_in[16];
  const float* dec_Whh   = (const float*)d_in[17];
  const float* dec_bih   = (const float*)d_in[18];
  const float* dec_bhh   = (const float*)d_in[19];
  const float* W_out     = (const float*)d_in[20];
  const float* b_out     = (const float*)d_in[21];
  const float* eps       = (const float*)d_in[22];

  float* ws = (float*)d_ws;
  float* enc_x  = ws;                       // [T,H]
  float* dec_x  = enc_x  + T_SZ * H_SZ;     // [T,H]
  float* XihE   = dec_x  + T_SZ * H_SZ;     // [T,4H]
  float* XihD   = XihE   + T_SZ * G4H;      // [T,4H]
  float* Hmat   = XihD   + T_SZ * G4H;      // [T,H]
  float* h_enc  = Hmat   + T_SZ * H_SZ;     // [H]  (h0, then running/final enc h)
  float* latent = h_enc  + H_SZ;            // [L]
  float* dec_h0 = latent + L_SZ;            // [H]
  float* dec_hg = dec_h0 + H_SZ;            // [H]  (running decoder h)
  float* logits = dec_hg + H_SZ;            // [T,V]
  unsigned* cnts = (unsigned*)(logits + (size_t)T_SZ * V_SZ);  // [2] barrier counters

  float* out        = (float*)d_out;
  float* out_pred   = out;                                // [T]
  float* out_dist   = out + T_SZ;                         // [T,V]
  float* out_mean   = out_dist + (size_t)T_SZ * V_SZ;     // [L]
  float* out_logvar = out_mean + L_SZ;                    // [L]

  // 1. embeddings + encoder h0 + barrier counter reset
  vae_prepare<<<T_SZ, 256, 0, stream>>>(word, tense, tense_emb, enc_emb, dec_emb,
                                        enc_x, dec_x, h_enc, cnts);

  // 2. Xih GEMMs (WMMA f32): [64,1024] @ [4096,1024]^T + (bih+bhh)
  {
    dim3 grid(G4H / (32 * 8));   // 8 waves/block, 64x32 tile per wave
    wmma_gemm_xwt<<<grid, 256, 0, stream>>>(enc_x, enc_Wih, enc_bih, enc_bhh, XihE, G4H, H_SZ);
    wmma_gemm_xwt<<<grid, 256, 0, stream>>>(dec_x, dec_Wih, dec_bih, dec_bhh, XihD, G4H, H_SZ);
  }

  // 3. encoder recurrence (h0 in h_enc; final h ends up in h_enc)
  lstm_seq_dist<<<NB_REC, 1024, 0, stream>>>(enc_Whh, XihE, h_enc, h_enc, nullptr, cnts + 0);

  // 4. latent + dec_h0
  latent_kernel<<<1, L_SZ, 0, stream>>>(h_enc, W_mean, b_mean, W_var, b_var, eps,
                                        out_mean, out_logvar, latent);
  dec_init<<<1, H_SZ, 0, stream>>>(latent, tense_emb, tense, W_l2h, b_l2h, dec_h0);

  // 5. decoder recurrence, recording all hidden states
  lstm_seq_dist<<<NB_REC, 1024, 0, stream>>>(dec_Whh, XihD, dec_h0, dec_hg, Hmat, cnts + 1);

  // 6. logits GEMM (WMMA f32): [64,1024] @ [32000,1024]^T + b_out
  {
    dim3 grid(V_SZ / (32 * 8));
    wmma_gemm_xwt<<<grid, 256, 0, stream>>>(Hmat, W_out, b_out, nullptr, logits, V_SZ, H_SZ);
  }

  // 7. log_softmax + argmax
  logsoftmax_argmax<<<T_SZ, 1024, 0, stream>>>(logits, out_dist, out_pred);
}